// Renderer_65833258713641
// MI455X (gfx1250) — compile-verified
//
#include <hip/hip_runtime.h>
#include <hip/hip_bf16.h>

// ---------------- problem constants ----------------
#define BATCH 2
#define NPTS  500
#define NFACE 512
#define IMH   128
#define IMW   128
#define TEXH  256
#define TEXW  256
#define SIGMA 0.0003f
#define INV_SIGMA (1.0f / SIGMA)

// ws SoA array ids (per batch, each of length NFACE floats)
#define A_E0 0
#define A_F0 1
#define A_G0 2
#define A_E1 3
#define A_F1 4
#define A_G1 5
#define A_E2 6
#define A_F2 7
#define A_G2 8
#define A_INV 9
#define A_DEN 10
#define A_Z0 11
#define A_Z1 12
#define A_Z2 13
#define A_AX 14
#define A_AY 15
#define A_BX 16
#define A_BY 17
#define A_CX 18
#define A_CY 19
#define A_E0X 20
#define A_E0Y 21
#define A_IL0 22
#define A_E1X 23
#define A_E1Y 24
#define A_IL1 25
#define A_E2X 26
#define A_E2Y 27
#define A_IL2 28
#define NARR 29
#define FD_WORDS (NARR * NFACE)   // 14848 floats = 58 KB

typedef __attribute__((ext_vector_type(2))) float v2f;
typedef __attribute__((ext_vector_type(8))) float v8f;
typedef __attribute__((ext_vector_type(4))) unsigned int u32x4;
typedef __attribute__((ext_vector_type(8))) unsigned int u32x8;

// ---------------------------------------------------------------------------
// Kernel 1: per-face precompute + normal output
// ---------------------------------------------------------------------------
__global__ void face_setup_kernel(const float* __restrict__ points,
                                  const int* __restrict__ faces,
                                  float* __restrict__ ws,
                                  float* __restrict__ out_normals) {
    int idx = blockIdx.x * blockDim.x + threadIdx.x;
    if (idx >= BATCH * NFACE) return;
    int b = idx / NFACE;
    int f = idx - b * NFACE;

    int i0 = faces[f * 3 + 0];
    int i1 = faces[f * 3 + 1];
    int i2 = faces[f * 3 + 2];
    const float* pb = points + b * NPTS * 3;

    float ax = pb[i0 * 3 + 0], ay = pb[i0 * 3 + 1], az = pb[i0 * 3 + 2];
    float bx = pb[i1 * 3 + 0], by = pb[i1 * 3 + 1], bz = pb[i1 * 3 + 2];
    float cx = pb[i2 * 3 + 0], cy = pb[i2 * 3 + 1], cz = pb[i2 * 3 + 2];

    // normal = cross(p1-p0, p2-p0)
    float ux = bx - ax, uy = by - ay, uz = bz - az;
    float vx = cx - ax, vy = cy - ay, vz = cz - az;
    float nx = uy * vz - uz * vy;
    float ny = uz * vx - ux * vz;
    float nz = ux * vy - uy * vx;
    float nrm = sqrtf(nx * nx + ny * ny + nz * nz) + 1e-10f;
    out_normals[(b * NFACE + f) * 3 + 0] = nx / nrm;
    out_normals[(b * NFACE + f) * 3 + 1] = ny / nrm;
    out_normals[(b * NFACE + f) * 3 + 2] = nz / nrm;

    float den = nz;
    float dsafe = (fabsf(den) < 1e-10f) ? 1e-10f : den;
    float inv = 1.0f / dsafe;

    float* wsb = ws + (size_t)b * FD_WORDS;
    // edge-function affine coefficients: wX = e + px*fc + py*g
    wsb[A_E0 * NFACE + f] = bx * cy - by * cx;
    wsb[A_F0 * NFACE + f] = by - cy;
    wsb[A_G0 * NFACE + f] = cx - bx;
    wsb[A_E1 * NFACE + f] = cx * ay - cy * ax;
    wsb[A_F1 * NFACE + f] = cy - ay;
    wsb[A_G1 * NFACE + f] = ax - cx;
    wsb[A_E2 * NFACE + f] = ax * by - ay * bx;
    wsb[A_F2 * NFACE + f] = ay - by;
    wsb[A_G2 * NFACE + f] = bx - ax;
    wsb[A_INV * NFACE + f] = inv;
    wsb[A_DEN * NFACE + f] = den;
    wsb[A_Z0 * NFACE + f] = az;
    wsb[A_Z1 * NFACE + f] = bz;
    wsb[A_Z2 * NFACE + f] = cz;
    wsb[A_AX * NFACE + f] = ax;
    wsb[A_AY * NFACE + f] = ay;
    wsb[A_BX * NFACE + f] = bx;
    wsb[A_BY * NFACE + f] = by;
    wsb[A_CX * NFACE + f] = cx;
    wsb[A_CY * NFACE + f] = cy;
    // per-edge segment data: direction + 1/(len^2 + 1e-12)
    float e0x = bx - ax, e0y = by - ay;
    float e1x = cx - bx, e1y = cy - by;
    float e2x = ax - cx, e2y = ay - cy;
    wsb[A_E0X * NFACE + f] = e0x;
    wsb[A_E0Y * NFACE + f] = e0y;
    wsb[A_IL0 * NFACE + f] = 1.0f / (e0x * e0x + e0y * e0y + 1e-12f);
    wsb[A_E1X * NFACE + f] = e1x;
    wsb[A_E1Y * NFACE + f] = e1y;
    wsb[A_IL1 * NFACE + f] = 1.0f / (e1x * e1x + e1y * e1y + 1e-12f);
    wsb[A_E2X * NFACE + f] = e2x;
    wsb[A_E2Y * NFACE + f] = e2y;
    wsb[A_IL2 * NFACE + f] = 1.0f / (e2x * e2x + e2y * e2y + 1e-12f);
}

// clamped point-to-segment squared distance (origin a, direction e, il = 1/(|e|^2+eps))
__device__ __forceinline__ float edge_d2(float ax, float ay, float ex, float ey,
                                         float il, float px, float py) {
    float rx = px - ax, ry = py - ay;
    float t = __saturatef((rx * ex + ry * ey) * il);
    float dx = rx - t * ex;
    float dy = ry - t * ey;
    return dx * dx + dy * dy;
}

__device__ __forceinline__ float pixel_x(int x) {
    return ((float)x + 0.5f) / (float)IMW * 2.0f - 1.0f;
}

// ---------------------------------------------------------------------------
// Kernel 2: rasterize one image row per block; WMMA edge-function evaluation.
//   grid = BATCH*IMH blocks, 256 threads (8 waves); wave w owns pixels [16w,16w+16)
//   Face table staged to LDS with a 1-D TENSOR_LOAD_TO_LDS (TDM) descriptor,
//   backstopped by a cooperative copy.  Per 16-face chunk: 3x
//   v_wmma_f32_16x16x4_f32 computes w0n/w1n/w2n for 16 pixels x 16 faces.
// ---------------------------------------------------------------------------
__global__ void __launch_bounds__(256)
raster_kernel(const float* __restrict__ ws,
              const int* __restrict__ faces,
              const float* __restrict__ uv,
              const float* __restrict__ tex,
              float* __restrict__ imrender,
              float* __restrict__ improb) {
    __shared__ float sfd[FD_WORDS];  // 29*512*4 = 58 KB of face data

    int b = blockIdx.x / IMH;
    int y = blockIdx.x - b * IMH;
    const float* wsb = ws + (size_t)b * FD_WORDS;

    __builtin_prefetch(wsb, 0, 1);  // global_prefetch_b8

    // ---- TDM async stage: 1-D tensor (14848 x 4B) global -> LDS ----
    if ((threadIdx.x >> 5) == 0) {   // one wave issues the DMA (EXEC ignored by TDM)
        unsigned long long ga = (unsigned long long)(const void*)wsb;
        unsigned int lds_base = (unsigned int)(unsigned long long)(const void*)&sfd[0];
        u32x4 g0;
        g0[0] = 1u;                                    // count=1, is_restore=0
        g0[1] = lds_base;                              // lds_addr[31:0]
        g0[2] = (unsigned int)(ga & 0xFFFFFFFFu);      // global_addr[31:0]
        g0[3] = (unsigned int)((ga >> 32) & 0x01FFFFFFu) | 0x80000000u; // ga[56:32] | type=2
        u32x8 g1;
        g1[0] = 0x00020000u;                           // wg_mask=0, data_size=2 (4B)
        g1[1] = ((unsigned)FD_WORDS & 0xFFFFu) << 16;  // tensor_dim0[15:0] @ bits[63:48]
        g1[2] = ((unsigned)FD_WORDS >> 16);            // tensor_dim0[31:16] @ bits[79:64]
        g1[3] = ((unsigned)FD_WORDS & 0xFFFFu) << 16;  // tile_dim0 @ bits[127:112]
        g1[4] = 0u;                                    // tile_dim1=0, tile_dim2=0
        g1[5] = 0u;
        g1[6] = 0u;
        g1[7] = 0u;
        asm volatile("tensor_load_to_lds %0, %1" :: "s"(g0), "s"(g1) : "memory");
    }
    __builtin_amdgcn_s_wait_tensorcnt(0);              // s_wait_tensorcnt 0x0

    // cooperative backstop copy (same bytes; guarantees LDS contents)
    for (int i = threadIdx.x; i < FD_WORDS; i += 256) sfd[i] = wsb[i];
    __syncthreads();

    int wave = threadIdx.x >> 5;
    int lane = threadIdx.x & 31;
    int hi = lane >> 4;       // 0: rows 0-7 / K0,K1 half ; 1: rows 8-15 / K2,K3 half
    int col = lane & 15;      // face column within chunk (also pixel M for A layout)
    int xbase = wave * 16;

    float py = 1.0f - ((float)y + 0.5f) / (float)IMH * 2.0f;

    // A matrix (16x4, M=pixel, K=[1,px,py,0]):
    //   lanes 0-15 : VGPR0=K0(=1), VGPR1=K1(=px[M])
    //   lanes 16-31: VGPR0=K2(=py), VGPR1=K3(=0)
    v2f amat;
    amat.x = hi ? py : 1.0f;
    amat.y = hi ? 0.0f : pixel_x(xbase + col);

    // branchless B-matrix fill: per-lane LDS offsets chosen once, .y masked by 0/1
    int offB0 = (hi ? A_G0 : A_E0) * NFACE;
    int offB1 = (hi ? A_G1 : A_E1) * NFACE;
    int offB2 = (hi ? A_G2 : A_E2) * NFACE;
    float loMask = hi ? 0.0f : 1.0f;

    float bestz[8], bw0[8], bw1[8], bw2[8], prodv[8];
    int bestf[8];
#pragma unroll
    for (int r = 0; r < 8; ++r) {
        bestz[r] = -1e30f;
        bw0[r] = bw1[r] = bw2[r] = 0.0f;
        prodv[r] = 1.0f;
        bestf[r] = 0;
    }

    for (int fb = 0; fb < NFACE; fb += 16) {
        int n = fb + col;
        v2f b0, b1, b2;
        b0.x = sfd[offB0 + n];
        b0.y = loMask * sfd[A_F0 * NFACE + n];
        b1.x = sfd[offB1 + n];
        b1.y = loMask * sfd[A_F1 * NFACE + n];
        b2.x = sfd[offB2 + n];
        b2.y = loMask * sfd[A_F2 * NFACE + n];

        v8f czero = {};
        v8f w0n = __builtin_amdgcn_wmma_f32_16x16x4_f32(false, amat, false, b0,
                                                        (short)0, czero, false, false);
        v8f w1n = __builtin_amdgcn_wmma_f32_16x16x4_f32(false, amat, false, b1,
                                                        (short)0, czero, false, false);
        v8f w2n = __builtin_amdgcn_wmma_f32_16x16x4_f32(false, amat, false, b2,
                                                        (short)0, czero, false, false);

        // per-lane face scalars (lanes in both halves read same face n)
        float finv = sfd[A_INV * NFACE + n];
        float fden = sfd[A_DEN * NFACE + n];
        float z0 = sfd[A_Z0 * NFACE + n];
        float z1 = sfd[A_Z1 * NFACE + n];
        float z2 = sfd[A_Z2 * NFACE + n];
        float fax = sfd[A_AX * NFACE + n], fay = sfd[A_AY * NFACE + n];
        float fbx = sfd[A_BX * NFACE + n], fby = sfd[A_BY * NFACE + n];
        float fcx = sfd[A_CX * NFACE + n], fcy = sfd[A_CY * NFACE + n];
        float e0x = sfd[A_E0X * NFACE + n], e0y = sfd[A_E0Y * NFACE + n];
        float il0 = sfd[A_IL0 * NFACE + n];
        float e1x = sfd[A_E1X * NFACE + n], e1y = sfd[A_E1Y * NFACE + n];
        float il1 = sfd[A_IL1 * NFACE + n];
        float e2x = sfd[A_E2X * NFACE + n], e2y = sfd[A_E2Y * NFACE + n];
        float il2 = sfd[A_IL2 * NFACE + n];
        bool front = fden > 0.0f;

#pragma unroll
        for (int r = 0; r < 8; ++r) {
            // D layout: VGPR r <-> pixel row (r + 8*hi), lane%16 <-> face
            float pxr = pixel_x(xbase + r + 8 * hi);
            float w0 = w0n[r] * finv;
            float w1 = w1n[r] * finv;
            float w2 = w2n[r] * finv;
            bool inside = front && (w0 >= 0.0f) && (w1 >= 0.0f) && (w2 >= 0.0f);
            float z = w0 * z0 + w1 * z1 + w2 * z2;
            float zb = inside ? z : -1e10f;
            // branchless z-argmax update (v_cndmask chain, no EXEC branches)
            bool take = zb > bestz[r];
            bestz[r] = take ? zb : bestz[r];
            bestf[r] = take ? n : bestf[r];
            bw0[r] = take ? w0 : bw0[r];
            bw1[r] = take ? w1 : bw1[r];
            bw2[r] = take ? w2 : bw2[r];

            float da = edge_d2(fax, fay, e0x, e0y, il0, pxr, py);
            float db = edge_d2(fbx, fby, e1x, e1y, il1, pxr, py);
            float dc = edge_d2(fcx, fcy, e2x, e2y, il2, pxr, py);
            float d2 = fminf(da, fminf(db, dc));
            d2 = inside ? 0.0f : d2;
            prodv[r] *= 1.0f - __expf(-d2 * INV_SIGMA);
        }
    }

    // butterfly reduce over the 16 faces-lanes in each half (wave32: xor 1,2,4,8
    // never crosses the 16-lane half boundary)
#pragma unroll
    for (int m = 1; m < 16; m <<= 1) {
#pragma unroll
        for (int r = 0; r < 8; ++r) {
            float oz = __shfl_xor(bestz[r], m, 32);
            int of = __shfl_xor(bestf[r], m, 32);
            float o0 = __shfl_xor(bw0[r], m, 32);
            float o1 = __shfl_xor(bw1[r], m, 32);
            float o2 = __shfl_xor(bw2[r], m, 32);
            float op = __shfl_xor(prodv[r], m, 32);
            prodv[r] *= op;
            // argmax tie-break: smaller face index wins (matches jnp.argmax)
            bool take = (oz > bestz[r]) || (oz == bestz[r] && of < bestf[r]);
            bestz[r] = take ? oz : bestz[r];
            bestf[r] = take ? of : bestf[r];
            bw0[r] = take ? o0 : bw0[r];
            bw1[r] = take ? o1 : bw1[r];
            bw2[r] = take ? o2 : bw2[r];
        }
    }

    // 16 lanes (col<8 in each half) finish one pixel each: UV gather + bilinear tex
    if (col < 8) {
        int r = col;
        int x = xbase + r + 8 * hi;
        float covered = (bestz[r] > -1e9f) ? 1.0f : 0.0f;
        float tw0 = bw0[r], tw1 = bw1[r], tw2 = bw2[r];
        float hardmask = (tw0 + tw1 + tw2) * covered;

        int n = bestf[r];
        int i0 = faces[n * 3 + 0];
        int i1 = faces[n * 3 + 1];
        int i2 = faces[n * 3 + 2];
        const float* uvb = uv + b * NPTS * 2;
        float u = (tw0 * uvb[i0 * 2 + 0] + tw1 * uvb[i1 * 2 + 0] + tw2 * uvb[i2 * 2 + 0]) * covered;
        float v = (tw0 * uvb[i0 * 2 + 1] + tw1 * uvb[i1 * 2 + 1] + tw2 * uvb[i2 * 2 + 1]) * covered;
        u = __saturatef(u);
        v = __saturatef(v);
        float xf = u * (float)(TEXW - 1);
        float yf = (1.0f - v) * (float)(TEXH - 1);
        int tx0 = (int)floorf(xf);
        int ty0 = (int)floorf(yf);
        int tx1 = min(tx0 + 1, TEXW - 1);
        int ty1 = min(ty0 + 1, TEXH - 1);
        float wx = xf - (float)tx0;
        float wy = yf - (float)ty0;

        const float* texb = tex + (size_t)b * 3 * TEXH * TEXW;
        size_t opix = ((size_t)b * IMH * IMW + (size_t)y * IMW + x);
#pragma unroll
        for (int c = 0; c < 3; ++c) {
            const float* tc = texb + (size_t)c * TEXH * TEXW;
            float c00 = tc[ty0 * TEXW + tx0];
            float c01 = tc[ty0 * TEXW + tx1];
            float c10 = tc[ty1 * TEXW + tx0];
            float c11 = tc[ty1 * TEXW + tx1];
            float color = (1.0f - wy) * ((1.0f - wx) * c00 + wx * c01) +
                          wy * ((1.0f - wx) * c10 + wx * c11);
            imrender[opix * 3 + c] = color * hardmask;
        }
        improb[opix] = 1.0f - prodv[r];
    }
}

// ---------------------------------------------------------------------------
extern "C" void kernel_launch(void* const* d_in, const int* in_sizes, int n_in,
                              void* d_out, int out_size, void* d_ws, size_t ws_size,
                              hipStream_t stream) {
    const float* points = (const float*)d_in[0];  // (B,P,3)
    const float* uv     = (const float*)d_in[1];  // (B,P,2)
    const float* tex    = (const float*)d_in[2];  // (B,3,TH,TW)
    const int*   faces  = (const int*)d_in[3];    // (F,3)

    float* out = (float*)d_out;
    float* imrender = out;                                        // B*H*W*3
    float* improb   = out + (size_t)BATCH * IMH * IMW * 3;        // B*H*W
    float* normals  = improb + (size_t)BATCH * IMH * IMW;         // B*F*3
    float* ws = (float*)d_ws;                                     // NARR*B*F floats

    face_setup_kernel<<<(BATCH * NFACE + 255) / 256, 256, 0, stream>>>(
        points, faces, ws, normals);
    raster_kernel<<<BATCH * IMH, 256, 0, stream>>>(
        ws, faces, uv, tex, imrender, improb);
}